// TemporalLayer_23132693856829
// MI455X (gfx1250) — compile-verified
//
#include <hip/hip_runtime.h>
#include <hip/hip_bf16.h>
#include <math.h>

// ---------------- problem constants ----------------
#define BATCH    2
#define SEQLEN   4096
#define BL       (BATCH * SEQLEN)          // 8192
#define D_MODEL  512
#define D_IN_PROJ 1168                     // 1024 + 128 + 16
#define CONV_DIM 576
#define CONV2    1152                      // 2*CONV_DIM
#define D_INNER  1024
#define NHEADS   16
#define HH       8                         // heads per direction
#define HEADDIM  64
#define NSTATE   32
#define NCHUNK   32
#define CHUNK    128                       // NCHUNK*CHUNK = SEQLEN

typedef float v2f __attribute__((ext_vector_type(2)));
typedef float v8f __attribute__((ext_vector_type(8)));

// Native fp32 WMMA: D(16x16,f32) = A(16x4,f32) x B(4x16,f32) + C
__device__ __forceinline__ v8f wmma_f32_4(v2f a, v2f b, v8f c) {
    return __builtin_amdgcn_wmma_f32_16x16x4_f32(
        /*neg_a=*/false, a, /*neg_b=*/false, b,
        /*c_mod=*/(short)0, c, /*reuse_a=*/false, /*reuse_b=*/false);
}

// =====================================================================
// Kernel 1: xbcdt = u @ W_in   (M=8192, K=512, N=1168)
// Wave computes 64(M) x 16(N): 4 accumulators share each B fragment.
// Block = 4 waves stacked in M -> block tile 256 x 16.
// =====================================================================
__global__ void __launch_bounds__(128)
gemm_in(const float* __restrict__ U, const float* __restrict__ Win,
        float* __restrict__ out) {
    const int wv   = threadIdx.x >> 5;
    const int lane = threadIdx.x & 31;
    const int row  = lane & 15;           // M within tile (for A) / N (for B)
    const int koff = (lane >> 4) << 1;    // K-pair select
    const int m0   = (blockIdx.y * 16 + wv * 4) * 16;   // wave covers m0..m0+63
    const int n0   = blockIdx.x * 16;

    const float* ap = U   + (size_t)(m0 + row) * D_MODEL + koff;
    const float* bp = Win + (size_t)koff * D_IN_PROJ + n0 + row;

    v8f acc0 = {}, acc1 = {}, acc2 = {}, acc3 = {};
    for (int k = 0; k < D_MODEL; k += 4) {
        v2f b;
        b.x = bp[(size_t)k * D_IN_PROJ];
        b.y = bp[(size_t)k * D_IN_PROJ + D_IN_PROJ];
        v2f a0 = *(const v2f*)(ap + k);
        v2f a1 = *(const v2f*)(ap + 16 * D_MODEL + k);
        v2f a2 = *(const v2f*)(ap + 32 * D_MODEL + k);
        v2f a3 = *(const v2f*)(ap + 48 * D_MODEL + k);
        acc0 = wmma_f32_4(a0, b, acc0);
        acc1 = wmma_f32_4(a1, b, acc1);
        acc2 = wmma_f32_4(a2, b, acc2);
        acc3 = wmma_f32_4(a3, b, acc3);
    }
    const int mhi = (lane >> 4) << 3;
#pragma unroll
    for (int r = 0; r < 8; ++r) {
        const size_t mm = (size_t)(m0 + r + mhi);
        out[(mm     ) * D_IN_PROJ + n0 + row] = acc0[r];
        out[(mm + 16) * D_IN_PROJ + n0 + row] = acc1[r];
        out[(mm + 32) * D_IN_PROJ + n0 + row] = acc2[r];
        out[(mm + 48) * D_IN_PROJ + n0 + row] = acc3[r];
    }
}

// =====================================================================
// Kernel 2: depthwise causal conv (fwd) / anti-causal conv (bwd, which is
// the causal conv on the time-reversed signal, expressed in original time)
// + SiLU, plus dt = softplus(raw + dt_bias)
// =====================================================================
__global__ void __launch_bounds__(256)
conv_dt(const float* __restrict__ xbcdt,
        const float* __restrict__ wf, const float* __restrict__ bf,
        const float* __restrict__ wb, const float* __restrict__ bb,
        const float* __restrict__ dt_bias,
        float* __restrict__ xc, float* __restrict__ dtb) {
    const size_t bl = blockIdx.x;
    const int b = (int)(bl / SEQLEN);
    const int t = (int)(bl % SEQLEN);

    for (int c = threadIdx.x; c < CONV2; c += blockDim.x) {
        float acc;
        if (c < CONV_DIM) {                       // forward: uses t-3..t
            acc = bf[c];
#pragma unroll
            for (int k = 0; k < 4; ++k) {
                int tt = t - 3 + k;
                if (tt >= 0)
                    acc += xbcdt[((size_t)b * SEQLEN + tt) * D_IN_PROJ + c] * wf[c * 4 + k];
            }
        } else {                                  // backward: uses t..t+3
            int cc = c - CONV_DIM;
            acc = bb[cc];
#pragma unroll
            for (int k = 0; k < 4; ++k) {
                int tt = t + 3 - k;
                if (tt < SEQLEN)
                    acc += xbcdt[((size_t)b * SEQLEN + tt) * D_IN_PROJ + c] * wb[cc * 4 + k];
            }
        }
        float sig = 1.0f / (1.0f + expf(-acc));
        xc[bl * CONV2 + c] = acc * sig;           // SiLU
    }
    for (int h = threadIdx.x; h < NHEADS; h += blockDim.x) {
        float v = xbcdt[bl * D_IN_PROJ + CONV2 + h] + dt_bias[h];
        dtb[bl * NHEADS + h] = (v > 20.0f) ? v : log1pf(expf(v));
    }
}

// head -> channel bases inside xc rows
__device__ __forceinline__ void head_bases(int h, int& xbase, int& bbase, int& cbase) {
    if (h < HH) { xbase = h * HEADDIM;                 bbase = 512;  cbase = 544;  }
    else        { xbase = CONV_DIM + (h - HH) * HEADDIM; bbase = 1088; cbase = 1120; }
}

// =====================================================================
// Kernel 3 (scan pass 1): per-(b,head,chunk) local scan from zero state.
// 64 threads, thread p owns state row h[p][0..31] in registers.
// Writes y_local (incl. D*x skip), final local state S, and prod(dA).
// =====================================================================
__global__ void __launch_bounds__(64)
scan_chunk(const float* __restrict__ xc, const float* __restrict__ dtb,
           const float* __restrict__ A_log, const float* __restrict__ Dp,
           float* __restrict__ y, float* __restrict__ S, float* __restrict__ prodA) {
    const int chunk = blockIdx.x & (NCHUNK - 1);
    const int h     = (blockIdx.x >> 5) & 15;
    const int b     = blockIdx.x >> 9;
    const int p     = threadIdx.x;
    int xbase, bbase, cbase; head_bases(h, xbase, bbase, cbase);

    const float Ah = -expf(A_log[h]);
    const float Dh = Dp[h];
    float hst[NSTATE];
#pragma unroll
    for (int n = 0; n < NSTATE; ++n) hst[n] = 0.0f;
    float pA = 1.0f;

    __shared__ float sB[NSTATE], sC[NSTATE];

    for (int sl = 0; sl < CHUNK; ++sl) {
        const int s = chunk * CHUNK + sl;
        const int t = (h < HH) ? s : (SEQLEN - 1 - s);   // bwd heads run reversed
        const size_t rb = (size_t)b * SEQLEN + t;
        const float dtv = dtb[rb * NHEADS + h];
        const float dA  = expf(Ah * dtv);
        pA *= dA;
        if (p < NSTATE) sB[p] = xc[rb * CONV2 + bbase + p];
        else            sC[p - NSTATE] = xc[rb * CONV2 + cbase + (p - NSTATE)];
        __syncthreads();
        const float xp  = xc[rb * CONV2 + xbase + p];
        const float dtx = dtv * xp;
        float accy = 0.0f;
#pragma unroll
        for (int n = 0; n < NSTATE; ++n) {
            hst[n] = dA * hst[n] + dtx * sB[n];
            accy  += hst[n] * sC[n];
        }
        y[rb * D_INNER + h * HEADDIM + p] = accy + Dh * xp;
        __syncthreads();
    }
    const size_t sb = (((size_t)(b * NHEADS + h)) * NCHUNK + chunk) * (HEADDIM * NSTATE)
                    + (size_t)p * NSTATE;
#pragma unroll
    for (int n = 0; n < NSTATE; ++n) S[sb + n] = hst[n];
    if (p == 0) prodA[(b * NHEADS + h) * NCHUNK + chunk] = pA;
}

// =====================================================================
// Kernel 4 (scan pass 2): sequential combine over the 32 chunks.
// In place: S[chunk] is replaced by the true initial state of that chunk.
// h_init[c+1] = prodA[c] * h_init[c] + S_local[c]
// =====================================================================
__global__ void __launch_bounds__(64)
scan_combine(float* __restrict__ S, const float* __restrict__ prodA) {
    const int bh = blockIdx.x;       // 0..31
    const int p  = threadIdx.x;      // 0..63
    float hcur[NSTATE];
#pragma unroll
    for (int n = 0; n < NSTATE; ++n) hcur[n] = 0.0f;
    for (int c = 0; c < NCHUNK; ++c) {
        const size_t base = (((size_t)bh) * NCHUNK + c) * (HEADDIM * NSTATE) + (size_t)p * NSTATE;
        const float a = prodA[bh * NCHUNK + c];
#pragma unroll
        for (int n = 0; n < NSTATE; ++n) {
            float sl = S[base + n];
            S[base + n] = hcur[n];                 // becomes h_init for chunk c
            hcur[n] = a * hcur[n] + sl;
        }
    }
}

// =====================================================================
// Kernel 5 (scan pass 3): y[t] += cumA_t * <h_init[p,:], C_t>
// cumA_t = prod of dA within chunk up to and including t.
// =====================================================================
__global__ void __launch_bounds__(64)
scan_correct(const float* __restrict__ xc, const float* __restrict__ dtb,
             const float* __restrict__ A_log, const float* __restrict__ S,
             float* __restrict__ y) {
    const int chunk = blockIdx.x & (NCHUNK - 1);
    if (chunk == 0) return;                        // h_init == 0
    const int h = (blockIdx.x >> 5) & 15;
    const int b = blockIdx.x >> 9;
    const int p = threadIdx.x;
    int xbase, bbase, cbase; head_bases(h, xbase, bbase, cbase);
    (void)xbase; (void)bbase;

    float hi[NSTATE];
    const size_t sb = (((size_t)(b * NHEADS + h)) * NCHUNK + chunk) * (HEADDIM * NSTATE)
                    + (size_t)p * NSTATE;
#pragma unroll
    for (int n = 0; n < NSTATE; ++n) hi[n] = S[sb + n];

    const float Ah = -expf(A_log[h]);
    float cumA = 1.0f;
    __shared__ float sC[NSTATE];

    for (int sl = 0; sl < CHUNK; ++sl) {
        const int s = chunk * CHUNK + sl;
        const int t = (h < HH) ? s : (SEQLEN - 1 - s);
        const size_t rb = (size_t)b * SEQLEN + t;
        cumA *= expf(Ah * dtb[rb * NHEADS + h]);
        if (p < NSTATE) sC[p] = xc[rb * CONV2 + cbase + p];
        __syncthreads();
        float acc = 0.0f;
#pragma unroll
        for (int n = 0; n < NSTATE; ++n) acc += hi[n] * sC[n];
        y[rb * D_INNER + h * HEADDIM + p] += cumA * acc;
        __syncthreads();
    }
}

// =====================================================================
// Kernel 6: per-row LayerNorm statistics (mu, rstd) over 1024 channels
// =====================================================================
__global__ void __launch_bounds__(256)
ln_stats(const float* __restrict__ y, float* __restrict__ mu, float* __restrict__ rstd) {
    const size_t row = blockIdx.x;
    const int tid = threadIdx.x;
    const float* yr = y + row * D_INNER;
    float s = 0.0f, q = 0.0f;
    for (int i = tid; i < D_INNER; i += 256) { float v = yr[i]; s += v; q += v * v; }
    __shared__ float ss[256], sq[256];
    ss[tid] = s; sq[tid] = q; __syncthreads();
    for (int off = 128; off > 0; off >>= 1) {
        if (tid < off) { ss[tid] += ss[tid + off]; sq[tid] += sq[tid + off]; }
        __syncthreads();
    }
    if (tid == 0) {
        float m = ss[0] * (1.0f / D_INNER);
        float v = sq[0] * (1.0f / D_INNER) - m * m;
        mu[row] = m;
        rstd[row] = rsqrtf(v + 1e-5f);
    }
}

// =====================================================================
// Kernel 7: out = LN(y) @ W_out  (M=8192, K=1024, N=512)
// LayerNorm fused into the A-operand load; wave computes 64(M) x 16(N).
// =====================================================================
__global__ void __launch_bounds__(128)
gemm_out(const float* __restrict__ Y, const float* __restrict__ mu,
         const float* __restrict__ rstd, const float* __restrict__ lnw,
         const float* __restrict__ lnb, const float* __restrict__ Wout,
         float* __restrict__ out) {
    const int wv   = threadIdx.x >> 5;
    const int lane = threadIdx.x & 31;
    const int row  = lane & 15;
    const int koff = (lane >> 4) << 1;
    const int m0   = (blockIdx.y * 16 + wv * 4) * 16;   // wave covers m0..m0+63
    const int n0   = blockIdx.x * 16;

    const float mum0 = mu[m0 + row],        rsm0 = rstd[m0 + row];
    const float mum1 = mu[m0 + 16 + row],   rsm1 = rstd[m0 + 16 + row];
    const float mum2 = mu[m0 + 32 + row],   rsm2 = rstd[m0 + 32 + row];
    const float mum3 = mu[m0 + 48 + row],   rsm3 = rstd[m0 + 48 + row];

    const float* ap = Y    + (size_t)(m0 + row) * D_INNER + koff;
    const float* bp = Wout + (size_t)koff * D_MODEL + n0 + row;

    v8f acc0 = {}, acc1 = {}, acc2 = {}, acc3 = {};
    for (int k = 0; k < D_INNER; k += 4) {
        const float w0 = lnw[k + koff], w1 = lnw[k + koff + 1];
        const float z0 = lnb[k + koff], z1 = lnb[k + koff + 1];
        v2f b;
        b.x = bp[(size_t)k * D_MODEL];
        b.y = bp[(size_t)k * D_MODEL + D_MODEL];
        v2f y0 = *(const v2f*)(ap + k);
        v2f y1 = *(const v2f*)(ap + 16 * D_INNER + k);
        v2f y2 = *(const v2f*)(ap + 32 * D_INNER + k);
        v2f y3 = *(const v2f*)(ap + 48 * D_INNER + k);
        v2f a0, a1, a2, a3;
        a0.x = (y0.x - mum0) * rsm0 * w0 + z0;  a0.y = (y0.y - mum0) * rsm0 * w1 + z1;
        a1.x = (y1.x - mum1) * rsm1 * w0 + z0;  a1.y = (y1.y - mum1) * rsm1 * w1 + z1;
        a2.x = (y2.x - mum2) * rsm2 * w0 + z0;  a2.y = (y2.y - mum2) * rsm2 * w1 + z1;
        a3.x = (y3.x - mum3) * rsm3 * w0 + z0;  a3.y = (y3.y - mum3) * rsm3 * w1 + z1;
        acc0 = wmma_f32_4(a0, b, acc0);
        acc1 = wmma_f32_4(a1, b, acc1);
        acc2 = wmma_f32_4(a2, b, acc2);
        acc3 = wmma_f32_4(a3, b, acc3);
    }
    const int mhi = (lane >> 4) << 3;
#pragma unroll
    for (int r = 0; r < 8; ++r) {
        const size_t mm = (size_t)(m0 + r + mhi);
        out[(mm     ) * D_MODEL + n0 + row] = acc0[r];
        out[(mm + 16) * D_MODEL + n0 + row] = acc1[r];
        out[(mm + 32) * D_MODEL + n0 + row] = acc2[r];
        out[(mm + 48) * D_MODEL + n0 + row] = acc3[r];
    }
}

// =====================================================================
extern "C" void kernel_launch(void* const* d_in, const int* in_sizes, int n_in,
                              void* d_out, int out_size, void* d_ws, size_t ws_size,
                              hipStream_t stream) {
    const float* u        = (const float*)d_in[0];
    // d_in[1] = seq_idx (unused: single segment)
    const float* W_in     = (const float*)d_in[2];
    const float* conv_fw  = (const float*)d_in[3];
    const float* conv_fb  = (const float*)d_in[4];
    const float* conv_bw  = (const float*)d_in[5];
    const float* conv_bb  = (const float*)d_in[6];
    const float* dt_bias  = (const float*)d_in[7];
    const float* A_log    = (const float*)d_in[8];
    const float* D_param  = (const float*)d_in[9];
    const float* ln_w     = (const float*)d_in[10];
    const float* ln_b     = (const float*)d_in[11];
    const float* W_out    = (const float*)d_in[12];
    float* out            = (float*)d_out;

    // workspace layout (floats)
    float* ws = (float*)d_ws;
    size_t off = 0;
    float* xbcdt = ws + off; off += (size_t)BL * D_IN_PROJ;      // 9,568,256
    float* xc    = ws + off; off += (size_t)BL * CONV2;          // 9,437,184
    float* dtb   = ws + off; off += (size_t)BL * NHEADS;         //   131,072
    float* ybuf  = ws + off; off += (size_t)BL * D_INNER;        // 8,388,608
    float* Sbuf  = ws + off; off += (size_t)32 * NCHUNK * HEADDIM * NSTATE; // 2,097,152
    float* pAbuf = ws + off; off += (size_t)32 * NCHUNK;         //     1,024
    float* mubuf = ws + off; off += (size_t)BL;                  //     8,192
    float* rsbuf = ws + off; off += (size_t)BL;                  //     8,192

    // 1) in-projection GEMM (fp32 WMMA, 64x16 wave tiles)
    gemm_in<<<dim3(D_IN_PROJ / 16, BL / 256), 128, 0, stream>>>(u, W_in, xbcdt);
    // 2) conv + SiLU + softplus(dt)
    conv_dt<<<dim3(BL), 256, 0, stream>>>(xbcdt, conv_fw, conv_fb, conv_bw, conv_bb,
                                          dt_bias, xc, dtb);
    // 3-5) chunked parallel selective scan
    scan_chunk<<<dim3(BATCH * NHEADS * NCHUNK), 64, 0, stream>>>(
        xc, dtb, A_log, D_param, ybuf, Sbuf, pAbuf);
    scan_combine<<<dim3(BATCH * NHEADS), 64, 0, stream>>>(Sbuf, pAbuf);
    scan_correct<<<dim3(BATCH * NHEADS * NCHUNK), 64, 0, stream>>>(
        xc, dtb, A_log, Sbuf, ybuf);
    // 6) LayerNorm stats
    ln_stats<<<dim3(BL), 256, 0, stream>>>(ybuf, mubuf, rsbuf);
    // 7) out-projection GEMM with fused LayerNorm (fp32 WMMA, 64x16 wave tiles)
    gemm_out<<<dim3(D_MODEL / 16, BL / 256), 128, 0, stream>>>(
        ybuf, mubuf, rsbuf, ln_w, ln_b, W_out, out);

    (void)in_sizes; (void)n_in; (void)out_size; (void)ws_size;
}